// LSHSoftmax_12661563589045
// MI455X (gfx1250) — compile-verified
//
#include <hip/hip_runtime.h>
#include <cstdint>

// Problem constants (reference: B=1024, D=512, N=500000, S=32768)
#define DDIM    512
#define BM      128
#define BN      128
#define BK      32
#define LDT     36          // LDS row stride in floats: 16B-aligned float4 slots, conflict-free b64 frag reads
#define THREADS 256         // 8 wave32s
#define KCHUNKS (DDIM / BK) // 16

typedef float v2f __attribute__((ext_vector_type(2)));
typedef float v4f __attribute__((ext_vector_type(4)));
typedef float v8f __attribute__((ext_vector_type(8)));

#if __has_builtin(__builtin_amdgcn_global_load_async_to_lds_b128)
#define HAVE_ASYNC_LDS 1
// Builtin signature (from hipcc diagnostic): (int4 AS1* src, int4 AS3* dst, imm offset, imm cpol)
typedef int v4ig __attribute__((vector_size(16)));
typedef __attribute__((address_space(1))) v4ig gbl_v4i_t;
typedef __attribute__((address_space(3))) v4ig lds_v4i_t;
#else
#define HAVE_ASYNC_LDS 0
#endif

__device__ __forceinline__ void wait_async_zero() {
#if __has_builtin(__builtin_amdgcn_s_wait_asynccnt)
    __builtin_amdgcn_s_wait_asynccnt(0);
#else
    asm volatile("s_wait_asynccnt 0" ::: "memory");
#endif
}

#if HAVE_ASYNC_LDS
__device__ __forceinline__ void async_cp16(const float* g, float* l) {
    __builtin_amdgcn_global_load_async_to_lds_b128(
        (gbl_v4i_t*)g, (lds_v4i_t*)l, 0, 0);
}
#endif

__global__ __launch_bounds__(THREADS)
void lsh_sampled_gemm(const float* __restrict__ inputs,   // [Btot, DDIM]
                      const float* __restrict__ weight,   // [N, DDIM]
                      const float* __restrict__ bias,     // [N]
                      const int*   __restrict__ sample_ids, // [S]
                      float* __restrict__ out,            // [Btot, S]
                      int S)
{
    __shared__ float lds[2][(BM + BN) * LDT]; // A tile then B tile, double buffered
    __shared__ int   s_ids[BN];
    __shared__ float s_bias[BN];

    const int tid = threadIdx.x;
    const int bn  = blockIdx.x;   // tile along S (sampled classes)
    const int bm  = blockIdx.y;   // tile along batch

    // Gather sampled ids + biases for this N-tile
    if (tid < BN) {
        int id = sample_ids[bn * BN + tid];
        s_ids[tid]  = id;
        s_bias[tid] = bias[id];
    }
    __syncthreads();

    // Per-thread copy slots: 4 float4 for A tile + 4 float4 for B tile per K-chunk.
    // slot s: id = tid + 256*s; row r = id>>3 in [0,127]; quad q = (id&7)*4 in {0..28}
    int ldsOff[4];
    const float* aG[4];
    const float* bG[4];
#pragma unroll
    for (int s2 = 0; s2 < 4; ++s2) {
        int id = tid + THREADS * s2;
        int r  = id >> 3;
        int q  = (id & 7) * 4;
        ldsOff[s2] = r * LDT + q;
        aG[s2] = inputs + (size_t)(bm * BM + r) * DDIM + q;
        bG[s2] = weight + (size_t)s_ids[r] * DDIM + q;
    }

    // ---- prologue: stage K-chunk 0 into buffer 0 ----
#if HAVE_ASYNC_LDS
#pragma unroll
    for (int s2 = 0; s2 < 4; ++s2) {
        async_cp16(aG[s2], &lds[0][ldsOff[s2]]);
        async_cp16(bG[s2], &lds[0][BM * LDT + ldsOff[s2]]);
    }
    wait_async_zero();
    __syncthreads();
#else
    {
        v4f ra[4], rb[4];
#pragma unroll
        for (int s2 = 0; s2 < 4; ++s2) {
            ra[s2] = *(const v4f*)aG[s2];
            rb[s2] = *(const v4f*)bG[s2];
        }
#pragma unroll
        for (int s2 = 0; s2 < 4; ++s2) {
            *(v4f*)&lds[0][ldsOff[s2]]            = ra[s2];
            *(v4f*)&lds[0][BM * LDT + ldsOff[s2]] = rb[s2];
        }
        __syncthreads();
    }
#endif

    // Wave/lane decomposition: 8 waves as 2 (M) x 4 (N); wave tile 64x32 = 4x2 WMMA tiles
    const int wave  = tid >> 5;
    const int lane  = tid & 31;
    const int wm    = wave >> 2;   // 0..1
    const int wn    = wave & 3;    // 0..3
    const int lr    = lane & 15;
    const int lhalf = lane >> 4;   // 0: K0/K1 halves, 1: K2/K3 halves (16x4 fp32 A layout)

    v8f acc[4][2];
#pragma unroll
    for (int mi = 0; mi < 4; ++mi)
#pragma unroll
        for (int ni = 0; ni < 2; ++ni)
            acc[mi][ni] = (v8f){0.f, 0.f, 0.f, 0.f, 0.f, 0.f, 0.f, 0.f};

    for (int kc = 0; kc < KCHUNKS; ++kc) {
        const int cur = kc & 1;
        const int nxt = cur ^ 1;
        const int k0n = (kc + 1) * BK;

#if HAVE_ASYNC_LDS
        if (kc + 1 < KCHUNKS) {
#pragma unroll
            for (int s2 = 0; s2 < 4; ++s2) {
                async_cp16(aG[s2] + k0n, &lds[nxt][ldsOff[s2]]);
                async_cp16(bG[s2] + k0n, &lds[nxt][BM * LDT + ldsOff[s2]]);
            }
        }
#else
        v4f ra[4], rb[4];
        if (kc + 1 < KCHUNKS) {
#pragma unroll
            for (int s2 = 0; s2 < 4; ++s2) {
                ra[s2] = *(const v4f*)(aG[s2] + k0n);
                rb[s2] = *(const v4f*)(bG[s2] + k0n);
            }
        }
#endif

        // ---- compute on current buffer: BK/4 = 8 WMMA K-steps ----
        const float* Ab = &lds[cur][0];
        const float* Bb = &lds[cur][BM * LDT];
#pragma unroll
        for (int ks = 0; ks < BK; ks += 4) {
            v2f afrag[4], bfrag[2];
#pragma unroll
            for (int mi = 0; mi < 4; ++mi) {
                int row = wm * 64 + mi * 16 + lr;
                afrag[mi] = *(const v2f*)&Ab[row * LDT + ks + lhalf * 2];
            }
#pragma unroll
            for (int ni = 0; ni < 2; ++ni) {
                int col = wn * 32 + ni * 16 + lr;
                bfrag[ni] = *(const v2f*)&Bb[col * LDT + ks + lhalf * 2];
            }
#pragma unroll
            for (int mi = 0; mi < 4; ++mi)
#pragma unroll
                for (int ni = 0; ni < 2; ++ni)
                    acc[mi][ni] = __builtin_amdgcn_wmma_f32_16x16x4_f32(
                        /*neg_a=*/false, afrag[mi],
                        /*neg_b=*/false, bfrag[ni],
                        /*c_mod=*/(short)0, acc[mi][ni],
                        /*reuse_a=*/false, /*reuse_b=*/false);
        }

#if HAVE_ASYNC_LDS
        wait_async_zero();   // my async writes to lds[nxt] landed
        __syncthreads();     // everyone done reading lds[cur] + writing lds[nxt]
#else
        __syncthreads();     // done reading lds[cur]
        if (kc + 1 < KCHUNKS) {
#pragma unroll
            for (int s2 = 0; s2 < 4; ++s2) {
                *(v4f*)&lds[nxt][ldsOff[s2]]            = ra[s2];
                *(v4f*)&lds[nxt][BM * LDT + ldsOff[s2]] = rb[s2];
            }
        }
        __syncthreads();
#endif
    }

    // ---- epilogue: add bias, store C.  C layout: VGPR r -> row tileM + r + lhalf*8, col tileN + lr
#pragma unroll
    for (int ni = 0; ni < 2; ++ni) {
        int colLocal = wn * 32 + ni * 16 + lr;
        int gcol     = bn * BN + colLocal;
        float bv     = s_bias[colLocal];
#pragma unroll
        for (int mi = 0; mi < 4; ++mi) {
            int gm0 = bm * BM + wm * 64 + mi * 16 + lhalf * 8;
#pragma unroll
            for (int r = 0; r < 8; ++r) {
                out[(size_t)(gm0 + r) * S + gcol] = acc[mi][ni][r] + bv;
            }
        }
    }
}

extern "C" void kernel_launch(void* const* d_in, const int* in_sizes, int n_in,
                              void* d_out, int out_size, void* d_ws, size_t ws_size,
                              hipStream_t stream) {
    (void)n_in; (void)d_ws; (void)ws_size; (void)out_size;
    const float* inputs     = (const float*)d_in[0];
    const float* weight     = (const float*)d_in[1];
    const float* bias       = (const float*)d_in[2];
    const int*   sample_ids = (const int*)d_in[3];
    float* out = (float*)d_out;

    const int Btot = in_sizes[0] / DDIM;   // 1024
    const int S    = in_sizes[3];          // 32768

    dim3 grid(S / BN, Btot / BM);          // (256, 8)
    lsh_sampled_gemm<<<grid, THREADS, 0, stream>>>(inputs, weight, bias, sample_ids, out, S);
}